// RoiPoolingConv_28862180229695
// MI455X (gfx1250) — compile-verified
//
#include <hip/hip_runtime.h>
#include <hip/hip_bf16.h>
#include <stdint.h>

#define POOL 7
#define NUM_ROIS 512
#define IMG_H 256
#define IMG_W 256
#define CHN 512
#define WAVES_PER_BLOCK 4
#define CELLS (NUM_ROIS * POOL * POOL)   // 25088, divisible by 4

__device__ __forceinline__ int clampi(int v, int lo, int hi) {
    return v < lo ? lo : (v > hi ? hi : v);
}

__global__ __launch_bounds__(32 * WAVES_PER_BLOCK)
void roi_pool_async_kernel(const float* __restrict__ img,
                           const int* __restrict__ rois,
                           float* __restrict__ out)
{
    // Per-wave LDS slice: 4 corner rows x 512 channels x 4B = 8KB
    __shared__ __align__(16) float smem[WAVES_PER_BLOCK * 4 * CHN];

    const int lane = threadIdx.x & 31;
    const int wave = threadIdx.x >> 5;
    const int cell = blockIdx.x * WAVES_PER_BLOCK + wave;   // exact, no tail

    const int roi = cell / (POOL * POOL);
    const int p   = cell - roi * (POOL * POOL);
    const int py  = p / POOL;
    const int px  = p - py * POOL;

    // ROI params (wave-uniform within the wave)
    const int rx = rois[roi * 4 + 0];
    const int ry = rois[roi * 4 + 1];
    const int rw = rois[roi * 4 + 2];
    const int rh = rois[roi * 4 + 3];

    // Bilinear sample coordinates (matches reference math exactly)
    const float sy = (float)py * ((float)rh / (float)POOL);
    const float sx = (float)px * ((float)rw / (float)POOL);
    const int   y0 = (int)floorf(sy);
    const int   x0 = (int)floorf(sx);
    const float fy = sy - (float)y0;
    const float fx = sx - (float)x0;
    const int   y1 = min(y0 + 1, rh - 1);
    const int   x1 = min(x0 + 1, rw - 1);
    const int ya0 = clampi(ry + y0, 0, IMG_H - 1);
    const int ya1 = clampi(ry + y1, 0, IMG_H - 1);
    const int xa0 = clampi(rx + x0, 0, IMG_W - 1);
    const int xa1 = clampi(rx + x1, 0, IMG_W - 1);

    // 4 corner rows, each a contiguous 512-float (2KB) vector in NHWC
    const float* g00 = img + ((size_t)ya0 * IMG_W + xa0) * CHN;
    const float* g01 = img + ((size_t)ya0 * IMG_W + xa1) * CHN;
    const float* g10 = img + ((size_t)ya1 * IMG_W + xa0) * CHN;
    const float* g11 = img + ((size_t)ya1 * IMG_W + xa1) * CHN;
    const float* gsrc[4] = { g00, g01, g10, g11 };

    float* wbase = smem + wave * 4 * CHN;

    // Stage 4 x 2KB corner rows into this wave's LDS slice with the gfx1250
    // async DMA path: 16 x global_load_async_to_lds_b128 (32 lanes x 16B each).
    #pragma unroll
    for (int k = 0; k < 4; ++k) {
        #pragma unroll
        for (int j = 0; j < 4; ++j) {
            // generic pointer to LDS: low 32 bits == LDS byte offset (aperture scheme)
            uint32_t ldsaddr = (uint32_t)(uintptr_t)(wbase + k * CHN + j * 128 + lane * 4);
            const float* g = gsrc[k] + j * 128 + lane * 4;
            asm volatile("global_load_async_to_lds_b128 %0, %1, off"
                         :: "v"(ldsaddr), "v"(g)
                         : "memory");
        }
    }
    // Per-wave async counter: only this wave reads its slice, no barrier needed.
    asm volatile("s_wait_asynccnt 0" ::: "memory");

    float* obase = out + (size_t)cell * CHN;

    #pragma unroll
    for (int t = 0; t < 4; ++t) {
        const int c = t * 128 + lane * 4;   // 16 channels per lane, float4 granularity
        const float4 v00 = *(const float4*)(wbase + 0 * CHN + c);
        const float4 v01 = *(const float4*)(wbase + 1 * CHN + c);
        const float4 v10 = *(const float4*)(wbase + 2 * CHN + c);
        const float4 v11 = *(const float4*)(wbase + 3 * CHN + c);

        float4 r;
        {
            float top = v00.x + (v01.x - v00.x) * fx;
            float bot = v10.x + (v11.x - v10.x) * fx;
            r.x = top + (bot - top) * fy;
        }
        {
            float top = v00.y + (v01.y - v00.y) * fx;
            float bot = v10.y + (v11.y - v10.y) * fx;
            r.y = top + (bot - top) * fy;
        }
        {
            float top = v00.z + (v01.z - v00.z) * fx;
            float bot = v10.z + (v11.z - v10.z) * fx;
            r.z = top + (bot - top) * fy;
        }
        {
            float top = v00.w + (v01.w - v00.w) * fx;
            float bot = v10.w + (v11.w - v10.w) * fx;
            r.w = top + (bot - top) * fy;
        }
        *(float4*)(obase + c) = r;
    }
}

extern "C" void kernel_launch(void* const* d_in, const int* in_sizes, int n_in,
                              void* d_out, int out_size, void* d_ws, size_t ws_size,
                              hipStream_t stream) {
    const float* img  = (const float*)d_in[0];
    const int*   rois = (const int*)d_in[1];
    float*       out  = (float*)d_out;

    dim3 block(32 * WAVES_PER_BLOCK);
    dim3 grid(CELLS / WAVES_PER_BLOCK);   // 6272 blocks, exact
    roi_pool_async_kernel<<<grid, block, 0, stream>>>(img, rois, out);
}